// HGNN_layer_86397562127082
// MI455X (gfx1250) — compile-verified
//
#include <hip/hip_runtime.h>
#include <hip/hip_bf16.h>

typedef float v2f __attribute__((ext_vector_type(2)));
typedef float v8f __attribute__((ext_vector_type(8)));

// ---------------------------------------------------------------------------
// GEMM: C[M,256] = A[M,256] * B[256,256], all f32 row-major.
// Uses V_WMMA_F32_16X16X4_F32 (f32 inputs, f32 accum) -> matches f32 reference.
// Block = 256 threads = 8 waves; each wave computes one 16x16 output tile.
// blockIdx.x = column tile (0..15), blockIdx.y*8 + wave = row tile.
//
// Fragment layouts (CDNA5 ISA 7.12.2, 32-bit A 16x4 / 32-bit C 16x16):
//   A: lanes 0-15 -> M=lane,    VGPR{0,1} = K{0,1}
//      lanes16-31 -> M=lane-16, VGPR{0,1} = K{2,3}
//   B (4x16, mirrored with N in lane dim):
//      lanes 0-15 -> N=lane,    VGPR{0,1} = K{0,1}
//      lanes16-31 -> N=lane-16, VGPR{0,1} = K{2,3}
//   C: VGPR v: lanes 0-15 -> (M=v,   N=lane), lanes16-31 -> (M=v+8, N=lane-16)
// ---------------------------------------------------------------------------
__global__ __launch_bounds__(256) void hgnn_gemm256_wmma(
    const float* __restrict__ A, const float* __restrict__ B,
    float* __restrict__ C, int M) {
  const int lane = threadIdx.x & 31;
  const int wave = threadIdx.x >> 5;
  const int half = lane >> 4;   // 0 or 1
  const int lm   = lane & 15;

  const int rowTile = blockIdx.y * 8 + wave;
  const int row0 = rowTile * 16;
  if (row0 >= M) return;                 // uniform per wave
  const int col0 = blockIdx.x * 16;

  int ar = row0 + lm;                    // clamp loads for partial last tile
  if (ar >= M) ar = M - 1;
  const float* __restrict__ Arow = A + (size_t)ar * 256 + 2 * half;
  const float* __restrict__ Bcol = B + (size_t)(2 * half) * 256 + col0 + lm;

  v8f c = {};
#pragma unroll 8
  for (int k = 0; k < 256; k += 4) {
    v2f a, b;
    a.x = Arow[k + 0];                       // A[r][k + 2*half + 0]
    a.y = Arow[k + 1];                       // A[r][k + 2*half + 1]
    b.x = Bcol[(size_t)k * 256];             // B[k + 2*half + 0][col0+lm]
    b.y = Bcol[(size_t)k * 256 + 256];       // B[k + 2*half + 1][col0+lm]
    // 8 args: (neg_a, A, neg_b, B, c_mod, C, reuse_a, reuse_b)
    c = __builtin_amdgcn_wmma_f32_16x16x4_f32(
        false, a, false, b, (short)0, c, false, false);
  }

  // Store epilogue. Fast path (wave-uniform): full 16-row tile in range ->
  // 8 unguarded coalesced b32 stores with immediate offsets (v*1024 bytes).
  float* __restrict__ Crow = C + (size_t)(row0 + 8 * half) * 256 + col0 + lm;
  if (row0 + 16 <= M) {
#pragma unroll
    for (int v = 0; v < 8; ++v) {
      Crow[(size_t)v * 256] = c[v];
    }
  } else {
#pragma unroll
    for (int v = 0; v < 8; ++v) {
      const int r = row0 + v + 8 * half;
      if (r < M) C[(size_t)r * 256 + col0 + lm] = c[v];
    }
  }
}

// ---------------------------------------------------------------------------
// Gather + masked-uniform-softmax reduce (+ optional ReLU):
//   dst[g,c] = (relu?) sum_l w[g,l] * src[idx[g,l], c]
// where w[g,l] = 1/cnt for idx>0 slots (cnt = #slots with idx>0),
//       or uniformly 1/32 if cnt==0   (exact softmax(where(idx>0,1,-9e15))).
// One block per row g; 256 threads = 256 channels. idx is wave-uniform, so
// the zero-weight skip is a uniform branch (saves the 1KB row load).
// ---------------------------------------------------------------------------
__global__ __launch_bounds__(256) void hgnn_gather_reduce(
    const float* __restrict__ src,   // [R, 256]
    const int*   __restrict__ idxmat,// [G, 32]
    float*       __restrict__ dst,   // [G, 256]
    int applyRelu) {
  const int g = blockIdx.x;
  const int c = threadIdx.x;
  const int* __restrict__ idx = idxmat + (size_t)g * 32;

  int cnt = 0;
#pragma unroll
  for (int l = 0; l < 32; ++l) cnt += (idx[l] > 0) ? 1 : 0;
  const float wpos = (cnt > 0) ? (1.0f / (float)cnt) : (1.0f / 32.0f);
  const bool allPad = (cnt == 0);

  float acc = 0.0f;
#pragma unroll 8
  for (int l = 0; l < 32; ++l) {
    const int id = idx[l];
    if (allPad || id > 0) {
      acc += wpos * src[(size_t)id * 256 + c];
    }
  }
  if (applyRelu) acc = fmaxf(acc, 0.0f);
  dst[(size_t)g * 256 + c] = acc;
}

// ---------------------------------------------------------------------------
// Pipeline: x1 = x@W1 ; edge = relu(gather(x1,seq)) ; e1 = edge@W2 ;
//           out = gather(e1, useq)
// Workspace: [ x1 : N*256 f32 ][ edge : E*256 f32 ][ e1 : E*256 f32 ]
// ---------------------------------------------------------------------------
extern "C" void kernel_launch(void* const* d_in, const int* in_sizes, int n_in,
                              void* d_out, int out_size, void* d_ws, size_t ws_size,
                              hipStream_t stream) {
  const float* x    = (const float*)d_in[0];
  const int*   seq  = (const int*)d_in[1];
  const int*   useq = (const int*)d_in[2];
  const float* W1   = (const float*)d_in[3];
  const float* W2   = (const float*)d_in[4];
  float* out = (float*)d_out;

  const int N = in_sizes[0] / 256;  // 50000
  const int E = in_sizes[1] / 32;   // 25000

  char* ws = (char*)d_ws;
  float* x1   = (float*)ws;
  float* edge = (float*)(ws + (size_t)N * 256 * sizeof(float));
  float* e1   = (float*)(ws + ((size_t)N + (size_t)E) * 256 * sizeof(float));

  const dim3 blk(256);

  {
    const int rowTiles = (N + 15) / 16;
    dim3 grid(16, (rowTiles + 7) / 8);
    hgnn_gemm256_wmma<<<grid, blk, 0, stream>>>(x, W1, x1, N);
  }

  hgnn_gather_reduce<<<dim3(E), blk, 0, stream>>>(x1, seq, edge, 1);

  {
    const int rowTiles = (E + 15) / 16;
    dim3 grid(16, (rowTiles + 7) / 8);
    hgnn_gemm256_wmma<<<grid, blk, 0, stream>>>(edge, W2, e1, E);
  }

  hgnn_gather_reduce<<<dim3(N), blk, 0, stream>>>(e1, useq, out, 0);
}